// TransformerBlockQuantum_65481071409702
// MI455X (gfx1250) — compile-verified
//
#include <hip/hip_runtime.h>

typedef __attribute__((ext_vector_type(2))) float v2f;
typedef __attribute__((ext_vector_type(8))) float v8f;

#define EPSLN 1e-5f

__device__ __forceinline__ v8f wmma_f32(v2f a, v2f b, v8f c) {
  // D(16x16,f32) = A(16x4,f32) x B(4x16,f32) + C
  return __builtin_amdgcn_wmma_f32_16x16x4_f32(false, a, false, b, (short)0, c,
                                               false, false);
}

// Gather a B-fragment (K=4 chunk starting at feature K0) for N-subtile s from
// lane-private per-token feature arrays (token == laneId).
// B layout (f32 4x16): V_j lane l: K = j + 2*(l>=16), N = l&15.
template <int K0>
__device__ __forceinline__ v2f bfrag_lane(const float (&v)[8], int s, int lane) {
  const int lhi = lane >> 4;
  const int lm = lane & 15;
  const int src = (s << 4) | lm;
  float a0 = lhi ? v[K0 + 2] : v[K0 + 0];
  float a1 = lhi ? v[K0 + 3] : v[K0 + 1];
  v2f b;
  b.x = __shfl(a0, src, 32);
  b.y = __shfl(a1, src, 32);
  return b;
}

// Gather B-fragment (K chunk C, features 4C..4C+3) from the two 16-row halves
// of the FFN hidden activation kept in D layout (Dh[0]=feats 0-15, Dh[1]=16-31).
template <int C>
__device__ __forceinline__ v2f bfrag_h(const v8f (&Dh)[2], int lane) {
  const int lhi = lane >> 4;
  const int lm = lane & 15;
  constexpr int HH = C >> 2;        // which 16-row half fragment
  constexpr int RB = (C >> 1) & 1;  // source lane block (rows 0-7 vs 8-15)
  constexpr int R0 = (4 * C) & 7;   // base accumulator register
  float a0 = lhi ? Dh[HH][R0 + 2] : Dh[HH][R0 + 0];
  float a1 = lhi ? Dh[HH][R0 + 3] : Dh[HH][R0 + 1];
  const int src = (RB << 4) | lm;
  v2f b;
  b.x = __shfl(a0, src, 32);
  b.y = __shfl(a1, src, 32);
  return b;
}

// Extract this lane's token (token == laneId) from two D fragments
// (Da: tokens 0-15, Db: tokens 16-31); rows 0-7 hold the 8 features.
__device__ __forceinline__ void d_to_lane(const v8f &Da, const v8f &Db, int lane,
                                          float (&out)[8]) {
  const int lm = lane & 15;
#pragma unroll
  for (int r = 0; r < 8; ++r) {
    float hv = __shfl(Db[r], lm, 32);
    out[r] = (lane < 16) ? Da[r] : hv;
  }
}

__device__ __forceinline__ void layernorm8(const float (&y)[8], const float (&g)[8],
                                           const float (&b)[8], float (&o)[8]) {
  float m = 0.f;
#pragma unroll
  for (int r = 0; r < 8; ++r) m += y[r];
  m *= 0.125f;
  float v = 0.f;
#pragma unroll
  for (int r = 0; r < 8; ++r) {
    float d = y[r] - m;
    v += d * d;
  }
  float rs = rsqrtf(v * 0.125f + EPSLN);
#pragma unroll
  for (int r = 0; r < 8; ++r) o[r] = (y[r] - m) * rs * g[r] + b[r];
}

__global__ void __launch_bounds__(256) qtb_kernel(
    const float *__restrict__ x, const float *__restrict__ Wq,
    const float *__restrict__ bq, const float *__restrict__ th_attn,
    const float *__restrict__ Wc, const float *__restrict__ bc,
    const float *__restrict__ g1, const float *__restrict__ be1,
    const float *__restrict__ th_ffn, const float *__restrict__ W1,
    const float *__restrict__ b1, const float *__restrict__ W2,
    const float *__restrict__ b2, const float *__restrict__ g2,
    const float *__restrict__ be2, float *__restrict__ out, int ntiles) {
  const int lane = threadIdx.x & 31;
  const int lhi = lane >> 4;
  const int lm = lane & 15;
  const int wave = (int)((blockIdx.x * blockDim.x + threadIdx.x) >> 5);
  const int nwaves = (int)((gridDim.x * blockDim.x) >> 5);

  // ---- A fragments: transposed weights, M = output feature, K = input ----
  // A layout (f32 16x4): lane l: M = l&15; V0 holds K = k0 + 2*(l>=16), V1 = +1.
  v2f AWq[2], AWc[2], AW1[2][2], AW2[8];
#pragma unroll
  for (int c = 0; c < 2; ++c) {
    const int k = 4 * c + 2 * lhi;
    AWq[c].x = (lm < 8) ? Wq[k * 8 + lm] : 0.f;
    AWq[c].y = (lm < 8) ? Wq[(k + 1) * 8 + lm] : 0.f;
    AWc[c].x = (lm < 8) ? Wc[k * 8 + lm] : 0.f;
    AWc[c].y = (lm < 8) ? Wc[(k + 1) * 8 + lm] : 0.f;
#pragma unroll
    for (int hh = 0; hh < 2; ++hh) {
      AW1[hh][c].x = W1[k * 32 + hh * 16 + lm];
      AW1[hh][c].y = W1[(k + 1) * 32 + hh * 16 + lm];
    }
  }
#pragma unroll
  for (int c = 0; c < 8; ++c) {
    const int k = 4 * c + 2 * lhi;
    AW2[c].x = (lm < 8) ? W2[k * 8 + lm] : 0.f;
    AW2[c].y = (lm < 8) ? W2[(k + 1) * 8 + lm] : 0.f;
  }

  // ---- Bias C fragments (row = output feature, broadcast over columns) ----
  v8f Cq, Cc, C2, Cb1[2];
#pragma unroll
  for (int r = 0; r < 8; ++r) {
    Cq[r] = lhi ? 0.f : bq[r];
    Cc[r] = lhi ? 0.f : bc[r];
    C2[r] = lhi ? 0.f : b2[r];
    Cb1[0][r] = b1[lhi * 8 + r];
    Cb1[1][r] = b1[16 + lhi * 8 + r];
  }

  // ---- Uniform per-feature parameters ----
  float th[8], ctf[8], vg1[8], vb1[8], vg2[8], vb2[8];
#pragma unroll
  for (int r = 0; r < 8; ++r) {
    th[r] = th_attn[r];
    ctf[r] = __cosf(th_ffn[r]);
    vg1[r] = g1[r];
    vb1[r] = be1[r];
    vg2[r] = g2[r];
    vb2[r] = be2[r];
  }

  for (int tile = wave; tile < ntiles; tile += nwaves) {
    const long long base = (long long)tile * 32;  // token index
    const float *xt = x + base * 8;

    if (tile + nwaves < ntiles)
      __builtin_prefetch(x + (base + (long long)nwaves * 32) * 8 + lane * 8, 0, 0);

    // x as B fragments (K = feature rows, N = tokens), direct coalesced b64 loads
    v2f Bx[2][2];
#pragma unroll
    for (int s = 0; s < 2; ++s)
#pragma unroll
      for (int c = 0; c < 2; ++c) {
        const float *p = xt + (s * 16 + lm) * 8 + 4 * c + 2 * lhi;
        v2f t;
        t.x = p[0];
        t.y = p[1];
        Bx[s][c] = t;
      }

    // lane-private x (token == laneId), for the residual paths
    float xl[8];
    {
      const float4 a = *(const float4 *)(xt + lane * 8);
      const float4 b = *(const float4 *)(xt + lane * 8 + 4);
      xl[0] = a.x; xl[1] = a.y; xl[2] = a.z; xl[3] = a.w;
      xl[4] = b.x; xl[5] = b.y; xl[6] = b.z; xl[7] = b.w;
    }

    // ---- GEMM 1: q = Wq^T * x^T (+bq) ----
    v8f Dq[2];
#pragma unroll
    for (int s = 0; s < 2; ++s) {
      v8f acc = wmma_f32(AWq[0], Bx[s][0], Cq);
      Dq[s] = wmma_f32(AWq[1], Bx[s][1], acc);
    }
    float q[8];
    d_to_lane(Dq[0], Dq[1], lane, q);

    // ---- Quantum attention measurement: RX ring -> <Z> chain products ----
    float ca[8];
#pragma unroll
    for (int r = 0; r < 8; ++r) ca[r] = __cosf(q[r] + th[r]);
    float z[8];
    {
      float run = ca[0];
#pragma unroll
      for (int r = 1; r < 8; ++r) {
        run *= ca[r];
        z[r] = run;
      }
      float u = ca[1];
#pragma unroll
      for (int r = 2; r < 8; ++r) u *= ca[r];
      z[0] = u;
    }

    // ---- GEMM 2: attn = Wc^T * z^T (+bc) ----
    v8f Dat[2];
#pragma unroll
    for (int s = 0; s < 2; ++s) {
      v2f B0 = bfrag_lane<0>(z, s, lane);
      v2f B4 = bfrag_lane<4>(z, s, lane);
      v8f acc = wmma_f32(AWc[0], B0, Cc);
      Dat[s] = wmma_f32(AWc[1], B4, acc);
    }
    float at[8];
    d_to_lane(Dat[0], Dat[1], lane, at);

    // ---- Residual + LayerNorm 1 ----
    float y1[8], x1l[8];
#pragma unroll
    for (int r = 0; r < 8; ++r) y1[r] = xl[r] + at[r];
    layernorm8(y1, vg1, vb1, x1l);

    // ---- FFN quantum encoder: zf = cos(theta_ffn)*cos(x1) ----
    float zf[8];
#pragma unroll
    for (int r = 0; r < 8; ++r) zf[r] = ctf[r] * __cosf(x1l[r]);

    // ---- GEMM 3: h = relu(W1^T * zf^T + b1), 32 rows as two 16-row halves ----
    v8f Dh[2][2];
#pragma unroll
    for (int s = 0; s < 2; ++s) {
      v2f B0 = bfrag_lane<0>(zf, s, lane);
      v2f B4 = bfrag_lane<4>(zf, s, lane);
#pragma unroll
      for (int hh = 0; hh < 2; ++hh) {
        v8f acc = wmma_f32(AW1[hh][0], B0, Cb1[hh]);
        acc = wmma_f32(AW1[hh][1], B4, acc);
#pragma unroll
        for (int r = 0; r < 8; ++r) acc[r] = fmaxf(acc[r], 0.f);
        Dh[s][hh] = acc;
      }
    }

    // ---- GEMM 4: ffn = W2^T * h (+b2), K=32 in 8 chunks ----
    v8f Dff[2];
#pragma unroll
    for (int s = 0; s < 2; ++s) {
      v8f acc = C2;
      acc = wmma_f32(AW2[0], bfrag_h<0>(Dh[s], lane), acc);
      acc = wmma_f32(AW2[1], bfrag_h<1>(Dh[s], lane), acc);
      acc = wmma_f32(AW2[2], bfrag_h<2>(Dh[s], lane), acc);
      acc = wmma_f32(AW2[3], bfrag_h<3>(Dh[s], lane), acc);
      acc = wmma_f32(AW2[4], bfrag_h<4>(Dh[s], lane), acc);
      acc = wmma_f32(AW2[5], bfrag_h<5>(Dh[s], lane), acc);
      acc = wmma_f32(AW2[6], bfrag_h<6>(Dh[s], lane), acc);
      acc = wmma_f32(AW2[7], bfrag_h<7>(Dh[s], lane), acc);
      Dff[s] = acc;
    }
    float fo[8];
    d_to_lane(Dff[0], Dff[1], lane, fo);

    // ---- Residual + LayerNorm 2, store ----
    float y2[8], o[8];
#pragma unroll
    for (int r = 0; r < 8; ++r) y2[r] = x1l[r] + fo[r];
    layernorm8(y2, vg2, vb2, o);

    float *op = out + (base + lane) * 8;
    float4 s0 = {o[0], o[1], o[2], o[3]};
    float4 s1 = {o[4], o[5], o[6], o[7]};
    *(float4 *)op = s0;
    *(float4 *)(op + 4) = s1;
  }
}

extern "C" void kernel_launch(void *const *d_in, const int *in_sizes, int n_in,
                              void *d_out, int out_size, void *d_ws,
                              size_t ws_size, hipStream_t stream) {
  (void)n_in;
  (void)out_size;
  (void)d_ws;
  (void)ws_size;
  const float *x = (const float *)d_in[0];
  const float *Wq = (const float *)d_in[1];
  const float *bq = (const float *)d_in[2];
  const float *th_attn = (const float *)d_in[3];
  const float *Wc = (const float *)d_in[4];
  const float *bc = (const float *)d_in[5];
  const float *g1 = (const float *)d_in[6];
  const float *be1 = (const float *)d_in[7];
  const float *th_ffn = (const float *)d_in[8];
  const float *W1 = (const float *)d_in[9];
  const float *b1 = (const float *)d_in[10];
  const float *W2 = (const float *)d_in[11];
  const float *b2 = (const float *)d_in[12];
  const float *g2 = (const float *)d_in[13];
  const float *be2 = (const float *)d_in[14];

  const int n_tokens = in_sizes[0] / 8;
  const int ntiles = n_tokens / 32;  // 32 tokens per wave-tile

  const int block = 256;                 // 8 waves per WGP-resident group
  int blocks = (ntiles + 8 - 1) / 8;     // >= 1 tile per wave if possible
  if (blocks > 1024) blocks = 1024;      // 8192 waves, grid-stride over tiles
  if (blocks < 1) blocks = 1;

  qtb_kernel<<<blocks, block, 0, stream>>>(x, Wq, bq, th_attn, Wc, bc, g1, be1,
                                           th_ffn, W1, b1, W2, b2, g2, be2,
                                           (float *)d_out, ntiles);
}